// RGCN_6098853560662
// MI455X (gfx1250) — compile-verified
//
#include <hip/hip_runtime.h>

typedef float v2f __attribute__((ext_vector_type(2)));
typedef float v8f __attribute__((ext_vector_type(8)));

#define N_NODES_C 50000
#define N_EDGES_C 800000
#define IN_F_C    96
#define HID_F_C   96
#define OUT_F_C   64
#define NUM_BASES_C 16

// ---------------------------------------------------------------------------
// Combine basis decomposition: W[r=0] = sum_b comp[0,b] * basis[b]
// (reference uses relation 0 for every edge)
// ---------------------------------------------------------------------------
__global__ void rgcn_build_w(const float* __restrict__ basis1,
                             const float* __restrict__ comp1,
                             const float* __restrict__ basis2,
                             const float* __restrict__ comp2,
                             float* __restrict__ W1,   // [96*96]
                             float* __restrict__ W2) { // [96*64]
    int t = blockIdx.x * blockDim.x + threadIdx.x;
    if (t < IN_F_C * HID_F_C) {
        float acc = 0.f;
        #pragma unroll
        for (int b = 0; b < NUM_BASES_C; ++b)
            acc += comp1[b] * basis1[(size_t)b * IN_F_C * HID_F_C + t];
        W1[t] = acc;
    }
    if (t < HID_F_C * OUT_F_C) {
        float acc = 0.f;
        #pragma unroll
        for (int b = 0; b < NUM_BASES_C; ++b)
            acc += comp2[b] * basis2[(size_t)b * HID_F_C * OUT_F_C + t];
        W2[t] = acc;
    }
}

// ---------------------------------------------------------------------------
// C[M,NCOL] = A[M,96] * B[96,NCOL] (+ bias), fp32 via V_WMMA_F32_16X16X4_F32.
// 128 threads = 4 waves; each wave computes a 16 x NCOL stripe.
// B (the weight matrix, <=36KB) is staged into LDS once per block.
// ---------------------------------------------------------------------------
template <int NCOL>
__global__ void __launch_bounds__(128)
rgcn_gemm_wmma(const float* __restrict__ A,
               const float* __restrict__ B,
               const float* __restrict__ bias,   // may be null
               float* __restrict__ C,
               int M) {
    constexpr int K  = 96;
    constexpr int NT = NCOL / 16;

    __shared__ float Blds[K * NCOL];
    for (int i = threadIdx.x; i < K * NCOL; i += 128)
        Blds[i] = B[i];
    __syncthreads();

    const int lane    = threadIdx.x & 31;
    const int wave    = threadIdx.x >> 5;
    const int rowBase = blockIdx.x * 64 + wave * 16;
    const int half    = lane >> 4;     // 0: lanes 0-15, 1: lanes 16-31
    const int lrow    = lane & 15;

    v8f acc[NT] = {};                  // zero accumulators

    // A row for this lane's fragment (clamped; OOB rows never stored)
    int arow = rowBase + lrow;
    if (arow > M - 1) arow = M - 1;
    const float* Ap = A + (size_t)arow * K;

    for (int kk = 0; kk < K; kk += 4) {
        // A fragment: lanes 0-15 carry K=kk,kk+1 ; lanes 16-31 carry K=kk+2,kk+3
        v2f a;
        a.x = Ap[kk + half * 2 + 0];
        a.y = Ap[kk + half * 2 + 1];
        #pragma unroll
        for (int t = 0; t < NT; ++t) {
            const int col = t * 16 + lrow;
            v2f b;
            b.x = Blds[(kk + half * 2 + 0) * NCOL + col];
            b.y = Blds[(kk + half * 2 + 1) * NCOL + col];
            acc[t] = __builtin_amdgcn_wmma_f32_16x16x4_f32(
                /*neg_a=*/false, a, /*neg_b=*/false, b,
                /*c_mod=*/(short)0, acc[t],
                /*reuse_a=*/false, /*reuse_b=*/false);
        }
    }

    // Epilogue: VGPR r holds M = r (lanes 0-15) / M = r+8 (lanes 16-31)
    #pragma unroll
    for (int t = 0; t < NT; ++t) {
        const int col = t * 16 + lrow;
        const float bv = bias ? bias[col] : 0.f;
        #pragma unroll
        for (int r = 0; r < 8; ++r) {
            const int row = rowBase + half * 8 + r;
            if (row < M)
                C[(size_t)row * NCOL + col] = acc[t][r] + bv;
        }
    }
}

// ---------------------------------------------------------------------------
// Edge scatter: out[dst[e]] += xW[src[e]] * edge_norm[e]
// One thread per (edge, 4-float chunk); HW fp32 global atomics.
// ---------------------------------------------------------------------------
template <int F>
__global__ void rgcn_edge_scatter(const float* __restrict__ xW,
                                  const float* __restrict__ enorm,
                                  const int* __restrict__ src,
                                  const int* __restrict__ dst,
                                  float* __restrict__ out) {
    constexpr int CH = F / 4;
    long long tid = (long long)blockIdx.x * blockDim.x + threadIdx.x;
    if (tid >= (long long)N_EDGES_C * CH) return;
    const int e = (int)(tid / CH);
    const int c = (int)(tid % CH) * 4;

    const float w = enorm[e];
    const int   s = src[e];
    const int   d = dst[e];

    const float4 v = *reinterpret_cast<const float4*>(xW + (size_t)s * F + c);
    float* o = out + (size_t)d * F + c;
    unsafeAtomicAdd(o + 0, v.x * w);
    unsafeAtomicAdd(o + 1, v.y * w);
    unsafeAtomicAdd(o + 2, v.z * w);
    unsafeAtomicAdd(o + 3, v.w * w);
}

__global__ void rgcn_relu(float* __restrict__ x, int n) {
    int i = blockIdx.x * blockDim.x + threadIdx.x;
    if (i < n) x[i] = fmaxf(x[i], 0.f);
}

// ---------------------------------------------------------------------------
extern "C" void kernel_launch(void* const* d_in, const int* in_sizes, int n_in,
                              void* d_out, int out_size, void* d_ws, size_t ws_size,
                              hipStream_t stream) {
    const float* node_feat = (const float*)d_in[0];
    const float* edge_feat = (const float*)d_in[1];
    const int*   src       = (const int*)  d_in[2];
    const int*   dst       = (const int*)  d_in[3];
    const float* basis1    = (const float*)d_in[4];
    const float* comp1     = (const float*)d_in[5];
    const float* loop_w1   = (const float*)d_in[6];
    const float* bias1     = (const float*)d_in[7];
    const float* basis2    = (const float*)d_in[8];
    const float* comp2     = (const float*)d_in[9];
    const float* loop_w2   = (const float*)d_in[10];
    const float* bias2     = (const float*)d_in[11];
    float* out = (float*)d_out;

    // workspace layout (floats, 16B-aligned chunks)
    float* ws  = (float*)d_ws;
    float* W1  = ws;                                   //  96*96  =  9216
    float* W2  = W1 + IN_F_C * HID_F_C;                //  96*64  =  6144
    float* xW1 = W2 + HID_F_C * OUT_F_C;               //  50000*96
    float* h1  = xW1 + (size_t)N_NODES_C * HID_F_C;    //  50000*96
    float* xW2 = h1  + (size_t)N_NODES_C * HID_F_C;    //  50000*64

    // 1) basis combine
    rgcn_build_w<<<(IN_F_C * HID_F_C + 255) / 256, 256, 0, stream>>>(
        basis1, comp1, basis2, comp2, W1, W2);

    const int gemmBlocks = (N_NODES_C + 63) / 64;

    // 2) layer 1: edge-message transform + self-loop (with bias)
    rgcn_gemm_wmma<HID_F_C><<<gemmBlocks, 128, 0, stream>>>(
        node_feat, W1, nullptr, xW1, N_NODES_C);
    rgcn_gemm_wmma<HID_F_C><<<gemmBlocks, 128, 0, stream>>>(
        node_feat, loop_w1, bias1, h1, N_NODES_C);

    // 3) layer 1 edge scatter into h1
    {
        long long th = (long long)N_EDGES_C * (HID_F_C / 4);
        rgcn_edge_scatter<HID_F_C><<<(int)((th + 255) / 256), 256, 0, stream>>>(
            xW1, edge_feat, src, dst, h1);
    }

    // 4) ReLU
    rgcn_relu<<<(N_NODES_C * HID_F_C + 255) / 256, 256, 0, stream>>>(
        h1, N_NODES_C * HID_F_C);

    // 5) layer 2
    rgcn_gemm_wmma<OUT_F_C><<<gemmBlocks, 128, 0, stream>>>(
        h1, W2, nullptr, xW2, N_NODES_C);
    rgcn_gemm_wmma<OUT_F_C><<<gemmBlocks, 128, 0, stream>>>(
        h1, loop_w2, bias2, out, N_NODES_C);

    // 6) layer 2 edge scatter into out
    {
        long long th = (long long)N_EDGES_C * (OUT_F_C / 4);
        rgcn_edge_scatter<OUT_F_C><<<(int)((th + 255) / 256), 256, 0, stream>>>(
            xW2, edge_feat, src, dst, out);
    }
}